// NoConsolidationModel_77068893160211
// MI455X (gfx1250) — compile-verified
//
#include <hip/hip_runtime.h>

typedef __attribute__((ext_vector_type(2))) float v2f;
typedef __attribute__((ext_vector_type(8))) float v8f;

namespace {
constexpr int HIDDEN    = 64;   // hidden dim and vocab (output) dim
constexpr int SEQ_LEN   = 64;
constexpr int MEM_SLOTS = 8;
constexpr int NROWS     = 66;   // VOCAB_SIZE + 2 embedding rows
constexpr int K1        = 128;  // concat dim (W1 inner dim)
constexpr int TST       = 66;   // LDS row stride for T1/T2 (pad: avoid 16-way conflicts)
constexpr int BROW      = 80;   // packed-B row length in pairs (2*80 % 64 == 32 -> halves disjoint)
constexpr int TILE_M    = 16;   // samples per wave tile
constexpr int NWAVES    = 8;    // waves per block (wave32)
constexpr int BLOCK_M   = NWAVES * TILE_M; // 128 samples per block
}

// ---------------------------------------------------------------------------
// Pre-kernel: T1[t][n] = b1[n] + embed[t] . W1[n][0:64]
//             T2[t][n] = 0.125 * (embed[t] . W1[n][64:128])
// tab layout: [NROWS*64] T1 followed by [NROWS*64] T2.
// ---------------------------------------------------------------------------
__global__ void ncm_tables(const float* __restrict__ embed,
                           const float* __restrict__ W1,
                           const float* __restrict__ b1,
                           float* __restrict__ tab) {
  const int total = NROWS * HIDDEN;
  int idx = blockIdx.x * blockDim.x + threadIdx.x;
  if (idx >= 2 * total) return;
  const int which = (idx >= total) ? 1 : 0;   // 0 -> T1, 1 -> T2
  const int i = which ? idx - total : idx;
  const int t = i >> 6;        // embedding row
  const int n = i & 63;        // output neuron
  const float* e = embed + t * HIDDEN;
  const float* w = W1 + n * K1 + (which ? HIDDEN : 0);
  float acc = which ? 0.0f : b1[n];
#pragma unroll 8
  for (int k = 0; k < HIDDEN; ++k) acc += e[k] * w[k];
  tab[idx] = which ? acc * 0.125f : acc;
}

// ---------------------------------------------------------------------------
// Main kernel: per 16-sample wave tile,
//   z[m][n] = T1[q_m][n] + sum_s T2[w_{m,s}][n]   (built directly in WMMA A layout)
//   logits = relu(z) @ W2^T + b2    via V_WMMA_F32_16X16X4_F32
// B fragments are pre-packed in LDS so each frag is one aligned ds_load_b64.
// ---------------------------------------------------------------------------
template <bool USE_WS>
__global__ __launch_bounds__(NWAVES * 32)
void ncm_main(const int* __restrict__ seqs,
              const int* __restrict__ qtok,
              const float* __restrict__ embed,
              const float* __restrict__ W1,
              const float* __restrict__ b1,
              const float* __restrict__ W2,
              const float* __restrict__ b2,
              const float* __restrict__ tab,
              float* __restrict__ out,
              int nB) {
  __shared__ float sT1[NROWS * TST];
  __shared__ float sT2[NROWS * TST];
  __shared__ v2f   sW2F[32 * BROW];   // packed B frags: [(kc*2+h)*BROW + n] = (W2T[k0][n], W2T[k0+1][n])
  __shared__ float sB2[HIDDEN];

  const int tid = threadIdx.x;

  // ---- cooperative LDS fill ----
  if constexpr (USE_WS) {
    for (int i = tid; i < NROWS * HIDDEN; i += NWAVES * 32) {
      const int t = i >> 6, n = i & 63;
      sT1[t * TST + n] = tab[i];
      sT2[t * TST + n] = tab[NROWS * HIDDEN + i];
    }
  } else {
    // fallback: recompute token tables per block
    for (int i = tid; i < NROWS * HIDDEN; i += NWAVES * 32) {
      const int t = i >> 6, n = i & 63;
      const float* e = embed + t * HIDDEN;
      const float* w = W1 + n * K1;
      float s1 = b1[n], s2 = 0.0f;
#pragma unroll 8
      for (int k = 0; k < HIDDEN; ++k) { s1 += e[k] * w[k]; s2 += e[k] * w[HIDDEN + k]; }
      sT1[t * TST + n] = s1;
      sT2[t * TST + n] = s2 * 0.125f;
    }
  }
  // pack W2 into B-fragment order: pair (k0, k0+1) contiguous, k0 = 4*kc + 2*h
  for (int i = tid; i < 32 * HIDDEN; i += NWAVES * 32) {
    const int n  = i & 63;        // B column (output neuron)
    const int kh = i >> 6;        // (kc*2 + h) in [0,32)
    const int k0 = 2 * kh;        // = 4*kc + 2*h
    v2f val;
    val.x = W2[n * HIDDEN + k0];
    val.y = W2[n * HIDDEN + k0 + 1];
    sW2F[kh * BROW + n] = val;
  }
  if (tid < HIDDEN) sB2[tid] = b2[tid];
  __syncthreads();

  // ---- per-wave 16-sample tile ----
  const int wave = tid >> 5;
  const int lane = tid & 31;
  const int m    = lane & 15;   // sample row within tile; also N-index for B/C/D frags
  const int half = lane >> 4;   // K-half selector for A/B frags
  const int tile0 = blockIdx.x * BLOCK_M + wave * TILE_M;
  int s = tile0 + m;
  if (s >= nB) s = nB - 1;      // clamp (keeps EXEC all-ones for WMMA); stores guarded below

  const int q = qtok[s];
  // window tokens: cols 55..62 of this row; row base is 256B aligned, col 52 is 16B aligned
  const int4* wv = (const int4*)(seqs + s * SEQ_LEN + (SEQ_LEN - 12));
  const int4 wa = wv[0], wb = wv[1], wc = wv[2];   // cols 52..55, 56..59, 60..63
  int wt[MEM_SLOTS] = { wa.w, wb.x, wb.y, wb.z, wb.w, wc.x, wc.y, wc.z };

  // accumulators init = b2 broadcast (D layout: N = 16*t + m for every VGPR row)
  v8f acc[4];
#pragma unroll
  for (int t = 0; t < 4; ++t) {
    const float bv = sB2[16 * t + m];
#pragma unroll
    for (int r = 0; r < 8; ++r) acc[t][r] = bv;
  }

  // K loop over hidden dim (64) in chunks of 4 for V_WMMA_F32_16X16X4_F32
#pragma unroll
  for (int kc = 0; kc < 16; ++kc) {
    const int k0 = 4 * kc + 2 * half;           // this lane's K pair
    // build relu(z) A-fragment pair by table gather-sum (9 rows)
    const float* p1 = &sT1[q * TST + k0];
    float a0 = p1[0], a1 = p1[1];
#pragma unroll
    for (int i = 0; i < MEM_SLOTS; ++i) {
      const float* p2 = &sT2[wt[i] * TST + k0];
      a0 += p2[0];
      a1 += p2[1];
    }
    v2f a;
    a.x = fmaxf(a0, 0.0f);
    a.y = fmaxf(a1, 0.0f);
    const v2f* brow = &sW2F[(kc * 2 + half) * BROW + m];
#pragma unroll
    for (int t = 0; t < 4; ++t) {
      const v2f b = brow[16 * t];               // one aligned ds_load_b64 per fragment
      acc[t] = __builtin_amdgcn_wmma_f32_16x16x4_f32(
          false, a, false, b, (short)0, acc[t], false, false);
    }
  }

  // ---- store D: row = tile0 + r + 8*half, col = 16*t + m ----
  const int row0 = tile0 + 8 * half;
  if (tile0 + TILE_M <= nB) {
    // fast path: whole tile in range, no per-store guards
#pragma unroll
    for (int t = 0; t < 4; ++t) {
#pragma unroll
      for (int r = 0; r < 8; ++r)
        out[(row0 + r) * HIDDEN + 16 * t + m] = acc[t][r];
    }
  } else {
#pragma unroll
    for (int t = 0; t < 4; ++t) {
#pragma unroll
      for (int r = 0; r < 8; ++r) {
        const int row = row0 + r;
        if (row < nB) out[row * HIDDEN + 16 * t + m] = acc[t][r];
      }
    }
  }
}

// ---------------------------------------------------------------------------
extern "C" void kernel_launch(void* const* d_in, const int* in_sizes, int n_in,
                              void* d_out, int out_size, void* d_ws, size_t ws_size,
                              hipStream_t stream) {
  const int*   seqs  = (const int*)d_in[0];
  const int*   qtok  = (const int*)d_in[1];
  const float* embed = (const float*)d_in[2];
  const float* W1    = (const float*)d_in[3];
  const float* b1    = (const float*)d_in[4];
  const float* W2    = (const float*)d_in[5];
  const float* b2    = (const float*)d_in[6];
  float*       out   = (float*)d_out;

  const int nB   = in_sizes[1];                 // batch size (query_tok count)
  const int grid = (nB + BLOCK_M - 1) / BLOCK_M;

  const size_t tabBytes = (size_t)2 * NROWS * HIDDEN * sizeof(float);
  if (ws_size >= tabBytes) {
    float* tab = (float*)d_ws;
    const int tgrid = (2 * NROWS * HIDDEN + 255) / 256;
    ncm_tables<<<tgrid, 256, 0, stream>>>(embed, W1, b1, tab);
    ncm_main<true><<<grid, NWAVES * 32, 0, stream>>>(
        seqs, qtok, embed, W1, b1, W2, b2, tab, out, nB);
  } else {
    ncm_main<false><<<grid, NWAVES * 32, 0, stream>>>(
        seqs, qtok, embed, W1, b1, W2, b2, nullptr, out, nB);
  }
}